// PMMs_28518582845722
// MI455X (gfx1250) — compile-verified
//
#include <hip/hip_runtime.h>
#include <hip/hip_bf16.h>

typedef __attribute__((ext_vector_type(2))) float v2f;
typedef __attribute__((ext_vector_type(8))) float v8f;

#define KAPPA 20.0f
#define C_DIM 256
#define N_DIM 4096      // 64*64
#define J_DIM 128       // 16 groups * 8 (k padded 5->8)
#define G_DIM 16
#define K_PRO 5
#define PH 64
#define HW 1048576      // 1024*1024

// ---------------- downsample mask: bilinear 1024->64 samples exactly a 2x2 block, w=0.5
__global__ __launch_bounds__(256) void down_kernel(const int* __restrict__ fg, float* __restrict__ m_small) {
    int idx = blockIdx.x * 256 + threadIdx.x;      // 16*4096
    int g = idx >> 12, n = idx & 4095;
    int y = n >> 6, x = n & 63;
    int r0 = 16 * y + 7, c0 = 16 * x + 7;
    const int* p = fg + g * HW;
    int a0 = __builtin_nontemporal_load(&p[r0 * 1024 + c0]);
    int a1 = __builtin_nontemporal_load(&p[r0 * 1024 + c0 + 1]);
    int a2 = __builtin_nontemporal_load(&p[(r0 + 1) * 1024 + c0]);
    int a3 = __builtin_nontemporal_load(&p[(r0 + 1) * 1024 + c0 + 1]);
    float s = (float)((a0 != 0) + (a1 != 0) + (a2 != 0) + (a3 != 0));
    m_small[idx] = 0.25f * s;
}

// ---------------- init MuAll[c*128+j] from mu input (1,256,5), l2norm over c, zero pads
__global__ __launch_bounds__(256) void init_mu_kernel(const float* __restrict__ mu, float* __restrict__ MuAll) {
    __shared__ float red[256];
    int j = blockIdx.x;            // 0..127
    int c = threadIdx.x;           // 0..255
    int kk = j & 7;
    float v = (kk < K_PRO) ? mu[c * K_PRO + kk] : 0.0f;
    red[c] = v * v;
    __syncthreads();
    for (int s = 128; s > 0; s >>= 1) { if (c < s) red[c] += red[c + s]; __syncthreads(); }
    float nrm = sqrtf(red[0]);
    MuAll[c * J_DIM + j] = (kk < K_PRO) ? v / (1e-6f + nrm) : 0.0f;
}

// ---------------- GEMM1: S[n,j] = sum_c F[c,n]*MuAll[c,j]   (M=n,N=j,K=c) via v_wmma_f32_16x16x4_f32
__global__ __launch_bounds__(32) void gemm1_kernel(const float* __restrict__ F,
                                                   const float* __restrict__ MuAll,
                                                   float* __restrict__ s_buf) {
    int tile = blockIdx.x;            // 256 n-tiles * 8 j-tiles
    int jt = tile & 7, nt = tile >> 3;
    int lane = threadIdx.x, m = lane & 15, hi = lane >> 4;
    int nb = nt * 16, jb = jt * 16;
    v8f acc = {};
    for (int c0 = 0; c0 < C_DIM; c0 += 4) {
        int ka = c0 + hi * 2;
        v2f a, b;
        a.x = F[(ka + 0) * N_DIM + nb + m];
        a.y = F[(ka + 1) * N_DIM + nb + m];
        b.x = MuAll[(ka + 0) * J_DIM + jb + m];
        b.y = MuAll[(ka + 1) * J_DIM + jb + m];
        acc = __builtin_amdgcn_wmma_f32_16x16x4_f32(false, a, false, b, (short)0, acc, false, false);
    }
    for (int r = 0; r < 8; ++r) {
        int row = r + hi * 8;
        s_buf[(nb + row) * J_DIM + jb + m] = acc[r];
    }
}

// ---------------- softmax over k (5) of KAPPA*m_small*S, store z, accumulate colsum
__global__ __launch_bounds__(256) void softz_kernel(const float* __restrict__ s_buf,
                                                    const float* __restrict__ m_small,
                                                    float* __restrict__ z_buf,
                                                    float* __restrict__ colsum) {
    __shared__ float ls[K_PRO];
    int g = blockIdx.x >> 4;
    int n = ((blockIdx.x & 15) << 8) + threadIdx.x;
    if (threadIdx.x < K_PRO) ls[threadIdx.x] = 0.0f;
    __syncthreads();
    float ms = m_small[g * N_DIM + n];
    float t[K_PRO], mx = -1e30f;
    for (int k = 0; k < K_PRO; ++k) {
        t[k] = KAPPA * ms * s_buf[n * J_DIM + g * 8 + k];
        mx = fmaxf(mx, t[k]);
    }
    float e[K_PRO], sum = 0.0f;
    for (int k = 0; k < K_PRO; ++k) { e[k] = __expf(t[k] - mx); sum += e[k]; }
    float inv = 1.0f / sum;
    for (int k = 0; k < K_PRO; ++k) {
        float z = e[k] * inv;
        z_buf[n * J_DIM + g * 8 + k] = z;
        atomicAdd(&ls[k], z);
    }
    __syncthreads();
    if (threadIdx.x < K_PRO) atomicAdd(&colsum[g * 8 + threadIdx.x], ls[threadIdx.x]);
}

// ---------------- zfold[n,j] = z[n,j] * m_small[g,n] / (1e-6 + colsum[j]); pads -> 0
__global__ __launch_bounds__(256) void zfold_kernel(const float* __restrict__ z_buf,
                                                    const float* __restrict__ m_small,
                                                    const float* __restrict__ colsum,
                                                    float* __restrict__ zfold) {
    int idx = blockIdx.x * 256 + threadIdx.x;    // 4096*128
    int n = idx >> 7, j = idx & 127;
    int g = j >> 3, kk = j & 7;
    float v = 0.0f;
    if (kk < K_PRO)
        v = z_buf[idx] * m_small[g * N_DIM + n] / (1e-6f + colsum[j]);
    zfold[idx] = v;
}

// ---------------- GEMM2: MuAcc[c,j] += sum_n F[c,n]*zfold[n,j]  (M=c,N=j,K=n) K-split x8
__global__ __launch_bounds__(32) void gemm2_kernel(const float* __restrict__ F,
                                                   const float* __restrict__ zfold,
                                                   float* __restrict__ mu_acc) {
    int wid = blockIdx.x;            // 128 tiles * 8 k-splits
    int ks = wid & 7, tile = wid >> 3;
    int jt = tile & 7, ct = tile >> 3;
    int lane = threadIdx.x, m = lane & 15, hi = lane >> 4;
    int cb = ct * 16, jb = jt * 16;
    v8f acc = {};
    int n0b = ks * 512;
    for (int n0 = n0b; n0 < n0b + 512; n0 += 4) {
        int kk = n0 + hi * 2;
        v2f a, b;
        a.x = F[(cb + m) * N_DIM + kk + 0];
        a.y = F[(cb + m) * N_DIM + kk + 1];
        b.x = zfold[(kk + 0) * J_DIM + jb + m];
        b.y = zfold[(kk + 1) * J_DIM + jb + m];
        acc = __builtin_amdgcn_wmma_f32_16x16x4_f32(false, a, false, b, (short)0, acc, false, false);
    }
    for (int r = 0; r < 8; ++r) {
        int row = r + hi * 8;
        atomicAdd(&mu_acc[(cb + row) * J_DIM + jb + m], acc[r]);
    }
}

// ---------------- l2norm columns of MuAcc over c -> MuAll
__global__ __launch_bounds__(256) void norm_kernel(const float* __restrict__ mu_acc, float* __restrict__ MuAll) {
    __shared__ float red[256];
    int j = blockIdx.x;
    int c = threadIdx.x;
    int kk = j & 7;
    float v = (kk < K_PRO) ? mu_acc[c * J_DIM + j] : 0.0f;
    red[c] = v * v;
    __syncthreads();
    for (int s = 128; s > 0; s >>= 1) { if (c < s) red[c] += red[c + s]; __syncthreads(); }
    float nrm = sqrtf(red[0]);
    MuAll[c * J_DIM + j] = (kk < K_PRO) ? v / (1e-6f + nrm) : 0.0f;
}

// ---------------- sim_c[g,k,n] = m_small[g,n] * S_final[n, g*8+k]
__global__ __launch_bounds__(256) void sim_kernel(const float* __restrict__ s_buf,
                                                  const float* __restrict__ m_small,
                                                  float* __restrict__ sim_c) {
    int idx = blockIdx.x * 256 + threadIdx.x;    // 16*5*4096
    int n = idx & 4095;
    int gk = idx >> 12;
    int g = gk / K_PRO, k = gk - g * K_PRO;
    sim_c[idx] = m_small[g * N_DIM + n] * s_buf[n * J_DIM + g * 8 + k];
}

// ---------------- upsample 16x + argmax over k + masked NT write + point/visibility reductions
__global__ __launch_bounds__(256) void up_kernel(const float* __restrict__ sim_c,
                                                 const int* __restrict__ fg,
                                                 float* __restrict__ part,
                                                 unsigned long long* __restrict__ best,
                                                 int* __restrict__ visible) {
    __shared__ unsigned long long wred[8];       // one partial per wave32
    __shared__ int vis[K_PRO];
    int idx = blockIdx.x * 256 + threadIdx.x;    // 16*1024*1024
    int g = idx >> 20;
    int p = idx & (HW - 1);
    int Y = p >> 10, X = p & 1023;
    int lane = threadIdx.x & 31;
    int wave = threadIdx.x >> 5;

    float fy = fminf(fmaxf((Y + 0.5f) * 0.0625f - 0.5f, 0.0f), 63.0f);
    float fx = fminf(fmaxf((X + 0.5f) * 0.0625f - 0.5f, 0.0f), 63.0f);
    int y0 = (int)fy, x0 = (int)fx;
    int y1 = min(y0 + 1, 63), x1 = min(x0 + 1, 63);
    float wy = fy - (float)y0, wx = fx - (float)x0;
    float w00 = (1.0f - wy) * (1.0f - wx), w01 = (1.0f - wy) * wx;
    float w10 = wy * (1.0f - wx), w11 = wy * wx;
    int i00 = y0 * PH + x0, i01 = y0 * PH + x1, i10 = y1 * PH + x0, i11 = y1 * PH + x1;

    const float* sc = sim_c + g * K_PRO * N_DIM;
    float v[K_PRO];
    for (int k = 0; k < K_PRO; ++k) {
        const float* s = sc + k * N_DIM;
        v[k] = w00 * s[i00] + w01 * s[i01] + w10 * s[i10] + w11 * s[i11];
    }
    int kbest = 0; float vb = v[0];
    for (int k = 1; k < K_PRO; ++k) if (v[k] > vb) { vb = v[k]; kbest = k; }  // first max on ties
    bool fgb = __builtin_nontemporal_load(&fg[idx]) != 0;

    if (threadIdx.x < K_PRO) vis[threadIdx.x] = 0;
    __syncthreads();

    for (int k = 0; k < K_PRO; ++k) {
        bool assign = fgb && (k == kbest);
        float mv = assign ? v[k] : 0.0f;
        __builtin_nontemporal_store(mv, &part[(size_t)(g * K_PRO + k) * HW + p]); // streamed once
        if (assign) atomicOr(&vis[k], 1);
        // packed key: high = order-preserving float map (max value wins), low = ~idx (min row-major idx on tie)
        unsigned ub = __float_as_uint(mv);
        ub = (ub & 0x80000000u) ? ~ub : (ub | 0x80000000u);
        unsigned long long pk = ((unsigned long long)ub << 32) | (unsigned long long)(0xFFFFFFFFu - (unsigned)p);
        // wave32 cross-lane max (no LDS, no barriers)
        for (int off = 16; off > 0; off >>= 1) {
            unsigned long long o = __shfl_xor(pk, off, 32);
            if (o > pk) pk = o;
        }
        if (lane == 0) wred[wave] = pk;
        __syncthreads();
        if (threadIdx.x == 0) {
            unsigned long long bb = wred[0];
            for (int w = 1; w < 8; ++w) if (wred[w] > bb) bb = wred[w];
            atomicMax(&best[g * K_PRO + k], bb);
        }
        __syncthreads();
    }
    if (threadIdx.x < K_PRO && vis[threadIdx.x]) atomicOr(&visible[g * K_PRO + threadIdx.x], 1);
}

// ---------------- finalize semantic points + visible weights
__global__ __launch_bounds__(128) void pts_kernel(const unsigned long long* __restrict__ best,
                                                  const int* __restrict__ visible,
                                                  float* __restrict__ out_pts,
                                                  float* __restrict__ out_vis) {
    int i = threadIdx.x;
    if (i >= G_DIM * K_PRO) return;
    int vsb = visible[i];
    float x = -0.001f, y = -0.001f;
    if (vsb) {
        unsigned low = (unsigned)(best[i] & 0xFFFFFFFFull);
        unsigned pidx = 0xFFFFFFFFu - low;
        x = (float)(pidx & 1023u);
        y = (float)(pidx >> 10);
    }
    out_pts[i * 2 + 0] = x;
    out_pts[i * 2 + 1] = y;
    out_vis[i] = vsb ? 1.0f : 0.0f;
}

extern "C" void kernel_launch(void* const* d_in, const int* in_sizes, int n_in,
                              void* d_out, int out_size, void* d_ws, size_t ws_size,
                              hipStream_t stream) {
    const float* feature = (const float*)d_in[0];   // (256,64,64)
    const int*   fg_mask = (const int*)d_in[1];     // (16,1024,1024)
    // d_in[2] = bboxes (unused by reference)
    const float* mu      = (const float*)d_in[3];   // (1,256,5)

    float* ws = (float*)d_ws;
    float* m_small = ws;                         // 65536
    float* MuAll   = ws + 65536;                 // 32768
    float* s_buf   = ws + 98304;                 // 524288
    float* z_buf   = ws + 622592;                // 524288
    float* zfold   = ws + 1146880;               // 524288
    float* colsum  = ws + 1671168;               // 128
    float* mu_acc  = ws + 1671296;               // 32768
    float* sim_c   = ws + 1704064;               // 327680
    unsigned long long* best = (unsigned long long*)(ws + 2031744); // 80 (8B-aligned)
    int* visible   = (int*)(ws + 2031904);       // 80

    float* part    = (float*)d_out;                       // 16*5*1024*1024
    float* out_pts = part + (size_t)G_DIM * K_PRO * HW;   // 160
    float* out_vis = out_pts + G_DIM * K_PRO * 2;         // 80

    down_kernel<<<256, 256, 0, stream>>>(fg_mask, m_small);
    init_mu_kernel<<<J_DIM, 256, 0, stream>>>(mu, MuAll);

    for (int it = 0; it < 10; ++it) {
        gemm1_kernel<<<2048, 32, 0, stream>>>(feature, MuAll, s_buf);
        hipMemsetAsync(colsum, 0, J_DIM * sizeof(float), stream);
        softz_kernel<<<256, 256, 0, stream>>>(s_buf, m_small, z_buf, colsum);
        zfold_kernel<<<2048, 256, 0, stream>>>(z_buf, m_small, colsum, zfold);
        hipMemsetAsync(mu_acc, 0, C_DIM * J_DIM * sizeof(float), stream);
        gemm2_kernel<<<1024, 32, 0, stream>>>(feature, zfold, mu_acc);
        norm_kernel<<<J_DIM, 256, 0, stream>>>(mu_acc, MuAll);
    }

    gemm1_kernel<<<2048, 32, 0, stream>>>(feature, MuAll, s_buf);   // final S with converged protos
    sim_kernel<<<1280, 256, 0, stream>>>(s_buf, m_small, sim_c);

    hipMemsetAsync(best, 0, G_DIM * K_PRO * sizeof(unsigned long long), stream);
    hipMemsetAsync(visible, 0, G_DIM * K_PRO * sizeof(int), stream);
    up_kernel<<<65536, 256, 0, stream>>>(sim_c, fg_mask, part, best, visible);
    pts_kernel<<<1, 128, 0, stream>>>(best, visible, out_pts, out_vis);
}